// HeadNNPU_30185030157000
// MI455X (gfx1250) — compile-verified
//
#include <hip/hip_runtime.h>

typedef __attribute__((ext_vector_type(2))) float v2f;
typedef __attribute__((ext_vector_type(8))) float v8f;
typedef int v4i __attribute__((vector_size(16)));                 // b128 payload
typedef __attribute__((address_space(3))) v4i as3v4i;             // LDS-side b128

#define KC 32            // K-chunk staged per iteration
#define TSTRIDE 36       // 32 + 4 pad floats; 36 % 4 == 0 keeps 16B alignment
#define ROWS_PER_BLOCK 64
#define THREADS 128      // 4 waves, each owns a 16-row tile
#define BUFSZ (ROWS_PER_BLOCK * TSTRIDE)

// ---------------------------------------------------------------------------
// Kernel 1: pred[r] = dot(z[r], W) + b  via V_WMMA_F32_16X16X4_F32 with a
// double-buffered GLOBAL_LOAD_ASYNC_TO_LDS_B128 pipeline (ASYNCcnt tracked).
// Memory-bound: 2B*D*4 = 256 MB of z at 23.3 TB/s ~= 11 us floor; the async
// pipeline keeps chunk i+1's HBM loads in flight under chunk i's WMMAs.
// ---------------------------------------------------------------------------
__global__ __launch_bounds__(THREADS)
void nnpu_gemv_wmma(const float* __restrict__ z_i,
                    const float* __restrict__ z_j,
                    const float* __restrict__ W,
                    const float* __restrict__ bias_p,
                    float* __restrict__ pred,
                    int B, int D) {
  __shared__ float tile[2 * BUFSZ];

  const int tid     = threadIdx.x;
  const int lane    = tid & 31;
  const int wave    = tid >> 5;
  const int laneLow = lane & 15;
  const int half    = lane >> 4;            // 0: K0/K1 lanes, 1: K2/K3 lanes
  const int rowBase = blockIdx.x * ROWS_PER_BLOCK;

  v8f acc = {};                              // 16x16 f32 D matrix (col 0 used)

  // issue one chunk's async loads into LDS buffer `buf`
  auto issue_chunk = [&](int kb, int buf) {
#pragma unroll
    for (int i = 0; i < (ROWS_PER_BLOCK * KC / 4) / THREADS; ++i) {
      const int f    = i * THREADS + tid;   // 0..511 float4 slots
      const int row  = f >> 3;              // 8 float4 per row
      const int c4   = (f & 7) << 2;
      const int grow = rowBase + row;
      const float* src = (grow < B ? z_i + (size_t)grow * D
                                   : z_j + (size_t)(grow - B) * D) + kb + c4;
      float* dst = &tile[buf * BUFSZ + row * TSTRIDE + c4];
      __builtin_amdgcn_global_load_async_to_lds_b128(
          (v4i*)src, (as3v4i*)dst, /*offset=*/0, /*cpol=*/0);
    }
  };

  issue_chunk(0, 0);                         // prologue: chunk 0 -> buf 0

  const int nChunks = D / KC;
  for (int c = 0; c < nChunks; ++c) {
    const int cur = c & 1;
    // Overlap the next chunk's HBM loads under this chunk's WMMAs.
    // Async loads retire in order: waiting <=4 (the batch just issued)
    // guarantees the current chunk's 4 loads have landed in LDS.
    if ((c + 1) < nChunks) {
      issue_chunk((c + 1) * KC, cur ^ 1);
      __builtin_amdgcn_s_wait_asynccnt(4);
    } else {
      __builtin_amdgcn_s_wait_asynccnt(0);
    }
    __syncthreads();

    const int kb = cur * BUFSZ;
#pragma unroll
    for (int kk = 0; kk < KC; kk += 4) {
      const int aoff = half << 1;                    // 0 or 2
      // A layout (16x4 f32): lane L<16 -> row L,K=kk..kk+1; lane16+L -> K=kk+2..+3
      const v2f a =
          *(const v2f*)&tile[kb + (wave * 16 + laneLow) * TSTRIDE + kk + aoff];
      // B (4x16 f32): W values only in column 0, other columns zero.
      const int widx = c * KC + kk + aoff;
      v2f bmat;
      bmat.x = (laneLow == 0) ? W[widx]     : 0.0f;  // branchless: EXEC stays full
      bmat.y = (laneLow == 0) ? W[widx + 1] : 0.0f;
      acc = __builtin_amdgcn_wmma_f32_16x16x4_f32(
          /*neg_a=*/false, a, /*neg_b=*/false, bmat,
          /*c_mod=*/(short)0, acc, /*reuse_a=*/false, /*reuse_b=*/false);
    }
    __syncthreads();   // all waves done with buf[cur] before it is re-targeted
  }

  // D column 0: lane 0 holds rows m=0..7 in acc[0..7], lane 16 rows m=8..15.
  if (laneLow == 0) {
    const float bv = bias_p[0];
    const int r0 = rowBase + wave * 16 + half * 8;
#pragma unroll
    for (int j = 0; j < 8; ++j) pred[r0 + j] = acc[j] + bv;
  }
}

// ---------------------------------------------------------------------------
// Kernel 2: total = sum(sigmoid(pred)), deterministic tree reduction.
// ---------------------------------------------------------------------------
__global__ __launch_bounds__(1024)
void nnpu_sigsum(const float* __restrict__ pred, float* __restrict__ total,
                 int n) {
  __shared__ float red[1024];
  const int t = threadIdx.x;
  float s = 0.0f;
  for (int i = t; i < n; i += 1024) {
    s += 1.0f / (1.0f + expf(-pred[i]));
  }
  red[t] = s;
  __syncthreads();
  for (int off = 512; off > 0; off >>= 1) {
    if (t < off) red[t] += red[t + off];
    __syncthreads();
  }
  if (t == 0) total[0] = red[0];
}

// ---------------------------------------------------------------------------
// Kernel 3: per-row nnPU risk + mean.  PRIOR=0.3, PRIOR'=0.5, GAMMA=1, BETA=0.
// ---------------------------------------------------------------------------
__global__ __launch_bounds__(1024)
void nnpu_final(const float* __restrict__ pred, const float* __restrict__ total_p,
                float* __restrict__ out, int n) {
  __shared__ float red[1024];
  const int t  = threadIdx.x;
  const int nh = n >> 1;
  const float total = total_p[0];
  const float c_unl = 0.5f / (0.7f * (float)(n - 2));   // (1-p')/((n-2)(1-p))
  const float c_inv = 0.15f / 1.4f;                     // (1-p')*p/(2*(1-p))
  float s = 0.0f;
  for (int i = t; i < n; i += 1024) {
    const float p  = pred[i];
    const int  pr  = (i < nh) ? (i + nh) : (i - nh);
    const float q  = pred[pr];
    const float sp  = 1.0f / (1.0f + expf(-p));
    const float sn  = 1.0f / (1.0f + expf(p));
    const float spq = 1.0f / (1.0f + expf(-q));
    const float snq = 1.0f / (1.0f + expf(q));
    const float sum_y_pos     = sn + snq;
    const float sum_y_pos_inv = sp + spq;
    const float sum_y_unl     = total - sp - spq;
    const float pos_risk = 0.25f * sum_y_pos;           // p'/2 * sum_y_pos
    const float neg_risk = c_unl * sum_y_unl - c_inv * sum_y_pos_inv;
    s += (neg_risk < 0.0f) ? -neg_risk : (pos_risk + neg_risk);
  }
  red[t] = s;
  __syncthreads();
  for (int off = 512; off > 0; off >>= 1) {
    if (t < off) red[t] += red[t + off];
    __syncthreads();
  }
  if (t == 0) out[0] = red[0] / (float)n;
}

extern "C" void kernel_launch(void* const* d_in, const int* in_sizes, int n_in,
                              void* d_out, int out_size, void* d_ws,
                              size_t ws_size, hipStream_t stream) {
  (void)n_in; (void)out_size; (void)ws_size;
  const float* z_i  = (const float*)d_in[0];
  const float* z_j  = (const float*)d_in[1];
  const float* W    = (const float*)d_in[2];
  const float* bias = (const float*)d_in[3];

  const int D = in_sizes[2];          // latent size (W length)
  const int B = in_sizes[0] / D;      // rows per half
  const int n = 2 * B;                // total pred rows

  float* pred  = (float*)d_ws;        // n floats
  float* total = pred + n;            // 1 float

  nnpu_gemv_wmma<<<n / ROWS_PER_BLOCK, THREADS, 0, stream>>>(
      z_i, z_j, W, bias, pred, B, D);
  nnpu_sigsum<<<1, 1024, 0, stream>>>(pred, total, n);
  nnpu_final<<<1, 1024, 0, stream>>>(pred, total, (float*)d_out, n);
}